// Net_6maxFull_36155034697770
// MI455X (gfx1250) — compile-verified
//
#include <hip/hip_runtime.h>
#include <hip/hip_bf16.h>

typedef __attribute__((ext_vector_type(2))) float v2f;
typedef __attribute__((ext_vector_type(8))) float v8f;

__device__ __forceinline__ float sigm(float v) { return 1.0f / (1.0f + expf(-v)); }

#define W1_LDS_STRIDE 204   // 204 % 64 = 12 -> conflict-free b64 reads across 16 lanes

// One workgroup, 256 threads (8 wave32). Whole net fused; intermediates in LDS.
__global__ __launch_bounds__(256, 1)
void net6max_fused_kernel(const float* __restrict__ x,
                          const float* __restrict__ gen_Wih,  // (10,40,12)
                          const float* __restrict__ gen_Whh,  // (10,40,10)
                          const float* __restrict__ gen_bih,  // (10,40)
                          const float* __restrict__ gen_bhh,  // (10,40)
                          const float* __restrict__ gen_h0,   // (10,10)
                          const float* __restrict__ gen_c0,   // (10,10)
                          const float* __restrict__ game_Wih, // (5,10,20,4)
                          const float* __restrict__ game_Whh, // (5,10,20,5)
                          const float* __restrict__ game_bih, // (5,10,20)
                          const float* __restrict__ game_bhh, // (5,10,20)
                          const float* __restrict__ game_h0,  // (5,10,5)
                          const float* __restrict__ game_c0,  // (5,10,5)
                          const float* __restrict__ W1,       // (50,200)
                          const float* __restrict__ b1,       // (50)
                          const float* __restrict__ W2,       // (10,50)
                          const float* __restrict__ b2,       // (10)
                          const float* __restrict__ W3,       // (1,10)
                          const float* __restrict__ b3,       // (1)
                          float* __restrict__ out)
{
    __shared__ float s_W1[64 * W1_LDS_STRIDE]; // padded W1 image (rows 50..63 zero)
    __shared__ float s_x[37];
    __shared__ float s_active[5];
    __shared__ float s_minv;           // 1 / mask.sum()
    __shared__ float s_gatesg[400];    // gen gates (10,40)
    __shared__ float s_gateso[1000];   // game gates (5,10,20)
    __shared__ float s_hgame[250];     // h_game (5,10,5)
    __shared__ float s_test[200];      // [0:100) x_gen_out, [100:200) opp_avg
    __shared__ float s_h1[64];         // layer-1 activations (50 valid)
    __shared__ float s_h2[10];

    const int tid = threadIdx.x;

    // ---- Stage 0: kick off async global->LDS copy of W1 (overlaps RNN stages) ----
    // 50 rows x 200 floats, copied in 16B chunks; tracked by ASYNCcnt.
    {
        const unsigned long long w1g = (unsigned long long)(uintptr_t)W1;
        for (int c = tid; c < 2500; c += 256) {           // 2500 chunks of 4 floats
            const int r  = c / 50;
            const int cc = (c % 50) * 4;
            const unsigned loff =
                (unsigned)(uintptr_t)(&s_W1[r * W1_LDS_STRIDE + cc]);
            const unsigned long long gaddr =
                w1g + (unsigned long long)((r * 200 + cc) * 4);
            asm volatile("global_load_async_to_lds_b128 %0, %1, off"
                         :: "v"(loff), "v"(gaddr) : "memory");
        }
        // zero-fill pad rows 50..63 so the WMMA B loads need no mask/branch
        for (int i = 50 * W1_LDS_STRIDE + tid; i < 64 * W1_LDS_STRIDE; i += 256)
            s_W1[i] = 0.0f;
    }

    // ---- Stage A: stage x into LDS ----
    if (tid < 37) s_x[tid] = x[tid];
    __syncthreads();

    if (tid < 5) s_active[tid] = s_x[12 + 5 * tid];
    __syncthreads();
    if (tid == 0) {
        float ms = 0.0f;
        #pragma unroll
        for (int o = 0; o < 5; ++o) ms += (s_active[o] == 1.0f) ? 1.0f : 0.0f;
        s_minv = 1.0f / ms;
    }

    // ---- Stage B: gen gates (10,40): Wih@xg + Whh@h0 + biases ----
    for (int t = tid; t < 400; t += 256) {
        const int k = t / 40;
        float acc = gen_bih[t] + gen_bhh[t];
        const float* wih = gen_Wih + t * 12;
        #pragma unroll
        for (int d = 0; d < 12; ++d) acc += wih[d] * s_x[d];
        const float* whh = gen_Whh + t * 10;
        const float* h0  = gen_h0 + k * 10;
        #pragma unroll
        for (int h = 0; h < 10; ++h) acc += whh[h] * h0[h];
        s_gatesg[t] = acc;
    }

    // ---- Stage C (overlapped): game gates (5,10,20): Wih@feats + Whh@h0 + biases ----
    for (int t = tid; t < 1000; t += 256) {
        const int o  = t / 200;
        const int kc = t / 20;          // (o*10+k)
        float acc = game_bih[t] + game_bhh[t];
        const float* wih  = game_Wih + t * 4;
        const float* feat = s_x + 12 + 5 * o + 1;   // feats[o][0:4]
        #pragma unroll
        for (int d = 0; d < 4; ++d) acc += wih[d] * feat[d];
        const float* whh = game_Whh + t * 5;
        const float* h0  = game_h0 + kc * 5;
        #pragma unroll
        for (int h = 0; h < 5; ++h) acc += whh[h] * h0[h];
        s_gateso[t] = acc;
    }
    __syncthreads();

    // ---- Stage D: gen LSTM cell -> test[0:100) ; game LSTM cell -> h_game ----
    if (tid < 100) {
        const int k = tid / 10, j = tid % 10, base = k * 40;
        const float gi = s_gatesg[base + j];
        const float gf = s_gatesg[base + 10 + j];
        const float gg = s_gatesg[base + 20 + j];
        const float go = s_gatesg[base + 30 + j];
        const float c  = sigm(gf) * gen_c0[tid] + sigm(gi) * tanhf(gg);
        s_test[tid] = sigm(go) * tanhf(c);
    } else if (tid < 100 + 250) {
        const int t = tid - 100;                 // (o,k,j) flat, j<5
        const int cell = t / 5, j = t % 5;       // cell = o*10+k
        const int base = cell * 20;
        const float gi = s_gateso[base + j];
        const float gf = s_gateso[base + 5 + j];
        const float gg = s_gateso[base + 10 + j];
        const float go = s_gateso[base + 15 + j];
        const float c  = sigm(gf) * game_c0[t] + sigm(gi) * tanhf(gg);
        s_hgame[t] = sigm(go) * tanhf(c);
    }
    __syncthreads();

    // ---- Stage E: masked opponent average -> test[100:200) ----
    if (tid < 100) {
        const int c = tid;
        const int idx = (c < 50) ? c : (45 + (c - 50) % 5);  // game_flat | repeated
        float acc = 0.0f;
        #pragma unroll
        for (int o = 0; o < 5; ++o) {
            const float m = (s_active[o] == 1.0f) ? 1.0f : 0.0f;
            acc += m * s_hgame[o * 50 + idx];
        }
        s_test[100 + c] = acc * s_minv;
    }

    // All waves drain their async W1 copies, then barrier makes LDS image
    // (plus s_test) visible to everyone.
    asm volatile("s_wait_asynccnt 0x0" ::: "memory");
    __syncthreads();

    // ---- Stage F: layer 1 via fp32 WMMA (v_wmma_f32_16x16x4_f32) ----
    // C(16x16) = A(16x4) x B(4x16) accumulated over K=200 in steps of 4.
    // A rows replicated with test[4k..4k+3] (broadcast LDS reads);
    // B[kk][n] = W1_lds[nt*16+n][4k+kk] (conflict-free ds_load_b64, zero-padded
    // rows >= 50 so the loop is branch-free with EXEC all-1s).
    {
        const int wave = tid >> 5;
        const int lane = tid & 31;
        if (wave < 4) {                       // wave-uniform branch
            const int nt   = wave;
            const int lrow = nt * 16 + (lane & 15);     // 0..63, padded rows are zero
            const int koff = (lane < 16) ? 0 : 2;
            const float* w1row = s_W1 + lrow * W1_LDS_STRIDE;
            v8f acc = {};
            #pragma unroll 5
            for (int k = 0; k < 50; ++k) {
                v2f a, b;
                a.x = s_test[4 * k + koff];
                a.y = s_test[4 * k + koff + 1];
                b.x = w1row[4 * k + koff];
                b.y = w1row[4 * k + koff + 1];
                acc = __builtin_amdgcn_wmma_f32_16x16x4_f32(
                        /*neg_a=*/false, a, /*neg_b=*/false, b,
                        /*c_mod=*/(short)0, acc,
                        /*reuse_a=*/false, /*reuse_b=*/false);
            }
            const int col = nt * 16 + lane;
            if (lane < 16 && col < 50) s_h1[col] = tanhf(acc[0] + b1[col]);
        }
    }
    __syncthreads();

    // ---- Stage G: layer 2 (10 x 50) ----
    if (tid < 10) {
        float acc = b2[tid];
        const float* w = W2 + tid * 50;
        #pragma unroll 10
        for (int j = 0; j < 50; ++j) acc += w[j] * s_h1[j];
        s_h2[tid] = tanhf(acc);
    }
    __syncthreads();

    // ---- Stage H: layer 3 (1 x 10) ----
    if (tid == 0) {
        float acc = b3[0];
        #pragma unroll
        for (int j = 0; j < 10; ++j) acc += W3[j] * s_h2[j];
        out[0] = tanhf(acc);
    }
}

extern "C" void kernel_launch(void* const* d_in, const int* in_sizes, int n_in,
                              void* d_out, int out_size, void* d_ws, size_t ws_size,
                              hipStream_t stream) {
    (void)in_sizes; (void)n_in; (void)out_size; (void)d_ws; (void)ws_size;
    net6max_fused_kernel<<<1, 256, 0, stream>>>(
        (const float*)d_in[0],  (const float*)d_in[1],  (const float*)d_in[2],
        (const float*)d_in[3],  (const float*)d_in[4],  (const float*)d_in[5],
        (const float*)d_in[6],  (const float*)d_in[7],  (const float*)d_in[8],
        (const float*)d_in[9],  (const float*)d_in[10], (const float*)d_in[11],
        (const float*)d_in[12], (const float*)d_in[13], (const float*)d_in[14],
        (const float*)d_in[15], (const float*)d_in[16], (const float*)d_in[17],
        (const float*)d_in[18], (float*)d_out);
}